// ShapeClassifier_86595130622246
// MI455X (gfx1250) — compile-verified
//
#include <hip/hip_runtime.h>
#include <math.h>

#define NPTS 1536
#define LN2F 0.6931471805599453f
#define GAMMA_RBF 1.1428571428571428f   /* 1/((3.5-0)/4) */
#define DCEN 1.1666666666666667f        /* 3.5/3 center spacing */
#define EPSF 1e-8f

typedef float v2f __attribute__((ext_vector_type(2)));
typedef float v8f __attribute__((ext_vector_type(8)));

struct Radial5 {
  const float* W1[5];
  const float* b1[5];
  const float* W2[5];
  const float* b2[5];
};

__device__ __forceinline__ float sspf(float x) {
  // shifted softplus, numerically stable
  return fmaxf(x, 0.f) + log1pf(expf(-fabsf(x))) - LN2F;
}

__device__ __forceinline__ v8f wmma_k4(v2f A, v2f B, v8f C) {
  // V_WMMA_F32_16X16X4_F32 : D = A(16x4) * B(4x16) + C(16x16)
  return __builtin_amdgcn_wmma_f32_16x16x4_f32(false, A, false, B, (short)0, C,
                                               false, false);
}

// DPP row_xmask butterfly add (stays within 16-lane rows; no LDS traffic)
template <int XOR>
__device__ __forceinline__ float dpp_xadd(float v) {
  int iv = __float_as_int(v);
  int sh = __builtin_amdgcn_update_dpp(0, iv, 0x160 | XOR, 0xf, 0xf, true);
  return v + __int_as_float(sh);
}
__device__ __forceinline__ float rowsum16(float v) {
  v = dpp_xadd<1>(v);
  v = dpp_xadd<2>(v);
  v = dpp_xadd<4>(v);
  v = dpp_xadd<8>(v);
  return v;
}

// ---------------------------------------------------------------------------
// Layer 0: Cin=1, paths 00 (scalar) and 01 (vector), in0 = embed scalar.
// ---------------------------------------------------------------------------
__global__ void layer0_msg(const float* __restrict__ x, const float* __restrict__ embed,
                           const float* __restrict__ W1a, const float* __restrict__ b1a,
                           const float* __restrict__ W2a, const float* __restrict__ b2a,
                           const float* __restrict__ W1b, const float* __restrict__ b1b,
                           const float* __restrict__ W2b, const float* __restrict__ b2b,
                           float* __restrict__ c0, float* __restrict__ c1) {
  int a = blockIdx.x * blockDim.x + threadIdx.x;
  if (a >= NPTS) return;
  int bs = blockIdx.y * 128, be = bs + 128;
  float xa0 = x[3*a], xa1 = x[3*a+1], xa2 = x[3*a+2];
  float e = embed[0];
  float w1A[16], w1B[16], b1A[4], b1B[4], w2A[4], w2B[4];
#pragma unroll
  for (int i = 0; i < 16; ++i) { w1A[i] = W1a[i]; w1B[i] = W1b[i]; }
#pragma unroll
  for (int i = 0; i < 4; ++i) { b1A[i] = b1a[i]; b1B[i] = b1b[i]; w2A[i] = W2a[i]; w2B[i] = W2b[i]; }
  float bb2A = b2a[0], bb2B = b2b[0];
  float m0 = 0.f, m1x = 0.f, m1y = 0.f, m1z = 0.f;
  for (int b = bs; b < be; ++b) {
    float r0 = xa0 - x[3*b], r1 = xa1 - x[3*b+1], r2 = xa2 - x[3*b+2];
    float q = r0*r0 + r1*r1 + r2*r2 + EPSF;
    float inv = __builtin_amdgcn_rsqf(q);   // 1/d since d = sqrt(q)
    float d = q * inv;
    float u0 = r0*inv, u1 = r1*inv, u2 = r2*inv;
    float rb[4];
#pragma unroll
    for (int k = 0; k < 4; ++k) { float t = d - DCEN*(float)k; rb[k] = __expf(-GAMMA_RBF*t*t); }
    float RA = bb2A, RB = bb2B;
#pragma unroll
    for (int h = 0; h < 4; ++h) {
      float hA = b1A[h], hB = b1B[h];
#pragma unroll
      for (int k = 0; k < 4; ++k) { hA += rb[k]*w1A[4*k+h]; hB += rb[k]*w1B[4*k+h]; }
      hA = fmaxf(hA, 0.f); hB = fmaxf(hB, 0.f);
      RA += hA*w2A[h]; RB += hB*w2B[h];
    }
    m0  += RA*e;
    m1x += RB*u0*e; m1y += RB*u1*e; m1z += RB*u2*e;
  }
  atomicAdd(c0 + a, m0);
  atomicAdd(c1 + 3*a + 0, m1x);
  atomicAdd(c1 + 3*a + 1, m1y);
  atomicAdd(c1 + 3*a + 2, m1z);
}

// ---------------------------------------------------------------------------
// Self-interaction + nonlinearity (per point). Cout = 4 always.
// ---------------------------------------------------------------------------
__global__ void self_interact(const float* __restrict__ cat0, const float* __restrict__ cat1,
                              const float* __restrict__ SI0W, const float* __restrict__ SI0b,
                              const float* __restrict__ SI1W, const float* __restrict__ NL1b,
                              float* __restrict__ f0o, float* __restrict__ f1o,
                              int n0C, int n1C) {
  int a = blockIdx.x * blockDim.x + threadIdx.x;
  if (a >= NPTS) return;
  float c0[8], c1[12][3];
  for (int f = 0; f < n0C; ++f) c0[f] = cat0[a*n0C + f];
  for (int f = 0; f < n1C; ++f) {
    int base = (a*n1C + f)*3;
    c1[f][0] = cat1[base]; c1[f][1] = cat1[base+1]; c1[f][2] = cat1[base+2];
  }
  for (int g = 0; g < 4; ++g) {
    float s = SI0b[g];
    for (int f = 0; f < n0C; ++f) s += SI0W[g*n0C + f]*c0[f];
    f0o[a*4 + g] = sspf(s);
    float v0 = 0.f, v1 = 0.f, v2 = 0.f;
    for (int f = 0; f < n1C; ++f) {
      float wv = SI1W[g*n1C + f];
      v0 += wv*c1[f][0]; v1 += wv*c1[f][1]; v2 += wv*c1[f][2];
    }
    float nrm = sqrtf(v0*v0 + v1*v1 + v2*v2 + EPSF);
    float fac = sspf(nrm + NL1b[g]) / nrm;
    f1o[(a*4+g)*3+0] = v0*fac;
    f1o[(a*4+g)*3+1] = v1*fac;
    f1o[(a*4+g)*3+2] = v2*fac;
  }
}

// ---------------------------------------------------------------------------
// Layers 1/2 message kernel: one wave per (a, b-chunk). 16 pairs per WMMA tile.
// 7 x V_WMMA_F32_16X16X4_F32 per tile, ZERO cross-lane ops in the loop:
//
// Stage-1 hidden row permutation: hidden unit (p,h) is produced at D row
//   d = 2p + (h&1) + 8*(h>>1)
// so D component 2p   = { h(p,0) | lanes0-15 , h(p,2) | lanes16-31 }  == B VGPR0
//    D component 2p+1 = { h(p,1)             , h(p,3)             }  == B VGPR1
// i.e. stage-2 B for path p is directly {h1[2p], h1[2p+1]} (B layout assumed:
// VGPR0 = K0 low / K2 high, VGPR1 = K1 low / K3 high).
// Path 4 hidden goes to D1B rows {0,1,8,9} -> B4 = {h1b[0], h1b[1]}.
// Stage-2 output rows m = 4p+c (block-diagonal A accumulation), unchanged.
// ---------------------------------------------------------------------------
__launch_bounds__(32)
__global__ void layer_msg_wmma(const float* __restrict__ x,
                               const float* __restrict__ f0in,   // [N*4]
                               const float* __restrict__ f1in,   // [N*4*3]
                               Radial5 P,
                               float* __restrict__ cat0,         // [N*8]
                               float* __restrict__ cat1) {       // [N*12*3]
  const int lane = threadIdx.x;
  const bool lo = lane < 16;
  const int a = blockIdx.x;
  const int m = lane & 15;
  const int k0 = lo ? 0 : 2;

  // ---- stage-1 A: row m holds weights for hidden unit (p(m), h(m)) ----
  const int p1m = (m & 7) >> 1;                 // path of stage-1 row m
  const int h1m = (m & 1) + ((m >> 3) << 1);    // hidden unit of row m
  v2f a1A, a1B;
  a1A[0] = P.W1[p1m][k0*4 + h1m];
  a1A[1] = P.W1[p1m][(k0+1)*4 + h1m];
  if ((m & 7) < 2) { a1B[0] = P.W1[4][k0*4 + h1m]; a1B[1] = P.W1[4][(k0+1)*4 + h1m]; }
  else             { a1B[0] = 0.f; a1B[1] = 0.f; }

  v8f c1A, c1B;
#pragma unroll
  for (int r = 0; r < 8; ++r) {
    int mm = r + (lo ? 0 : 8);
    int pp = (mm & 7) >> 1, hh = (mm & 1) + ((mm >> 3) << 1);
    c1A[r] = P.b1[pp][hh];
    c1B[r] = ((mm & 7) < 2) ? P.b1[4][hh] : 0.f;
  }

  // ---- stage-2 A: output rows m = 4p + c (block-diagonal gating) ----
  const int pm = m >> 2, fm = m & 3;
  float w2l = P.W2[pm][k0*4 + fm], w2h = P.W2[pm][(k0+1)*4 + fm];
  v2f a2p0 = { pm == 0 ? w2l : 0.f, pm == 0 ? w2h : 0.f };
  v2f a2p1 = { pm == 1 ? w2l : 0.f, pm == 1 ? w2h : 0.f };
  v2f a2p2 = { pm == 2 ? w2l : 0.f, pm == 2 ? w2h : 0.f };
  v2f a2p3 = { pm == 3 ? w2l : 0.f, pm == 3 ? w2h : 0.f };
  v2f a2p4;
  if (m < 4) { a2p4[0] = P.W2[4][k0*4 + m]; a2p4[1] = P.W2[4][(k0+1)*4 + m]; }
  else       { a2p4[0] = 0.f; a2p4[1] = 0.f; }

  v8f c2A, c2B;
#pragma unroll
  for (int r = 0; r < 8; ++r) {
    int mm = r + (lo ? 0 : 8);
    c2A[r] = P.b2[mm >> 2][mm & 3];
    c2B[r] = (mm < 4) ? P.b2[4][mm] : 0.f;
  }

  float xa0 = x[3*a], xa1 = x[3*a+1], xa2 = x[3*a+2];

  float accA[16], accB[12];
#pragma unroll
  for (int i = 0; i < 16; ++i) accA[i] = 0.f;
#pragma unroll
  for (int i = 0; i < 12; ++i) accB[i] = 0.f;

  const int bstart = blockIdx.y * 256;
  for (int bb = bstart; bb < bstart + 256; bb += 16) {
    int b = bb + m;                         // lanes l and l+16 share pair j = l%16
    float r0 = xa0 - x[3*b], r1 = xa1 - x[3*b+1], r2 = xa2 - x[3*b+2];
    float q = r0*r0 + r1*r1 + r2*r2 + EPSF;
    float inv = __builtin_amdgcn_rsqf(q);   // == 1/d, no IEEE divide chain
    float d = q * inv;
    float u0 = r0*inv, u1 = r1*inv, u2 = r2*inv;
    float t0 = d, t1 = d - DCEN, t2 = d - 2.f*DCEN, t3 = d - 3.f*DCEN;
    float e0 = __expf(-GAMMA_RBF*t0*t0), e1 = __expf(-GAMMA_RBF*t1*t1),
          e2 = __expf(-GAMMA_RBF*t2*t2), e3 = __expf(-GAMMA_RBF*t3*t3);

    // stage-1: hidden = W1*rbf + b1 for all 5 paths (permuted row order)
    v2f B1; B1[0] = lo ? e0 : e2; B1[1] = lo ? e1 : e3;
    v8f h1  = wmma_k4(a1A, B1, c1A);
    v8f h1b = wmma_k4(a1B, B1, c1B);
#pragma unroll
    for (int r = 0; r < 8; ++r) h1[r] = fmaxf(h1[r], 0.f);
    h1b[0] = fmaxf(h1b[0], 0.f);
    h1b[1] = fmaxf(h1b[1], 0.f);

    // stage-2: R = W2*relu(hidden) + b2; B operands come straight from D regs
    v8f RA = c2A, RB = c2B;
    { v2f B = { h1[0], h1[1] };   RA = wmma_k4(a2p0, B, RA); }   // path 00
    { v2f B = { h1[2], h1[3] };   RA = wmma_k4(a2p1, B, RA); }   // path 01
    { v2f B = { h1[4], h1[5] };   RA = wmma_k4(a2p2, B, RA); }   // path 10
    { v2f B = { h1[6], h1[7] };   RA = wmma_k4(a2p3, B, RA); }   // path 11f0
    { v2f B = { h1b[0], h1b[1] }; RB = wmma_k4(a2p4, B, RB); }   // path 11f1

    // neighbor features (small, L2-resident)
    float i0f[4], i1f[4][3];
#pragma unroll
    for (int f = 0; f < 4; ++f) i0f[f] = f0in[b*4 + f];
#pragma unroll
    for (int f = 0; f < 4; ++f) {
      i1f[f][0] = f1in[b*12 + f*3 + 0];
      i1f[f][1] = f1in[b*12 + f*3 + 1];
      i1f[f][2] = f1in[b*12 + f*3 + 2];
    }
    float RAv[8], RBv[4];
#pragma unroll
    for (int r = 0; r < 8; ++r) RAv[r] = RA[r];
#pragma unroll
    for (int r = 0; r < 4; ++r) RBv[r] = RB[r];

    if (lo) {  // rows 0-7: path00 (f=r), path01 (f=r-4); plus path11f1 from RB
#pragma unroll
      for (int f = 0; f < 4; ++f) accA[f] += RAv[f]*i0f[f];
#pragma unroll
      for (int f = 0; f < 4; ++f) {
        float s = RAv[4+f]*i0f[f];
        accA[4+3*f+0] += s*u0; accA[4+3*f+1] += s*u1; accA[4+3*f+2] += s*u2;
      }
#pragma unroll
      for (int f = 0; f < 4; ++f) {
        float cx = u1*i1f[f][2] - u2*i1f[f][1];
        float cy = u2*i1f[f][0] - u0*i1f[f][2];
        float cz = u0*i1f[f][1] - u1*i1f[f][0];
        float R = RBv[f];
        accB[3*f+0] += R*cx; accB[3*f+1] += R*cy; accB[3*f+2] += R*cz;
      }
    } else {   // rows 8-15: path10 (f=r), path11f0 (f=r-4)
#pragma unroll
      for (int f = 0; f < 4; ++f) {
        float dt = u0*i1f[f][0] + u1*i1f[f][1] + u2*i1f[f][2];
        accA[f] += RAv[f]*dt;
      }
#pragma unroll
      for (int f = 0; f < 4; ++f) {
        float R = RAv[4+f];
        accA[4+3*f+0] += R*i1f[f][0];
        accA[4+3*f+1] += R*i1f[f][1];
        accA[4+3*f+2] += R*i1f[f][2];
      }
    }
  }

  // sum over the 16 pair lanes within each half-wave (DPP row_xmask, no LDS)
#pragma unroll
  for (int i = 0; i < 16; ++i) accA[i] = rowsum16(accA[i]);
#pragma unroll
  for (int i = 0; i < 12; ++i) accB[i] = rowsum16(accB[i]);

  if (lane == 0) {
#pragma unroll
    for (int f = 0; f < 4; ++f) atomicAdd(cat0 + a*8 + f, accA[f]);          // path00
#pragma unroll
    for (int f = 0; f < 4; ++f) {
      atomicAdd(cat1 + (a*12 +     f)*3 + 0, accA[4+3*f+0]);                 // path01
      atomicAdd(cat1 + (a*12 +     f)*3 + 1, accA[4+3*f+1]);
      atomicAdd(cat1 + (a*12 +     f)*3 + 2, accA[4+3*f+2]);
      atomicAdd(cat1 + (a*12 + 8 + f)*3 + 0, accB[3*f+0]);                   // path11f1
      atomicAdd(cat1 + (a*12 + 8 + f)*3 + 1, accB[3*f+1]);
      atomicAdd(cat1 + (a*12 + 8 + f)*3 + 2, accB[3*f+2]);
    }
  } else if (lane == 16) {
#pragma unroll
    for (int f = 0; f < 4; ++f) atomicAdd(cat0 + a*8 + 4 + f, accA[f]);      // path10
#pragma unroll
    for (int f = 0; f < 4; ++f) {
      atomicAdd(cat1 + (a*12 + 4 + f)*3 + 0, accA[4+3*f+0]);                 // path11f0
      atomicAdd(cat1 + (a*12 + 4 + f)*3 + 1, accA[4+3*f+1]);
      atomicAdd(cat1 + (a*12 + 4 + f)*3 + 2, accA[4+3*f+2]);
    }
  }
}

// ---------------------------------------------------------------------------
// Head: mean-pool over points, then pooled @ W_cls^T + b_cls  -> out[8]
// ---------------------------------------------------------------------------
__global__ void head_kernel(const float* __restrict__ f0,
                            const float* __restrict__ Wc, const float* __restrict__ bc,
                            float* __restrict__ out) {
  __shared__ float red[256];
  __shared__ float pool[4];
  float s[4] = {0.f, 0.f, 0.f, 0.f};
  for (int a = threadIdx.x; a < NPTS; a += 256) {
    s[0] += f0[a*4+0]; s[1] += f0[a*4+1]; s[2] += f0[a*4+2]; s[3] += f0[a*4+3];
  }
  for (int g = 0; g < 4; ++g) {
    red[threadIdx.x] = s[g];
    __syncthreads();
    for (int st = 128; st > 0; st >>= 1) {
      if (threadIdx.x < st) red[threadIdx.x] += red[threadIdx.x + st];
      __syncthreads();
    }
    if (threadIdx.x == 0) pool[g] = red[0] * (1.0f / (float)NPTS);
    __syncthreads();
  }
  if (threadIdx.x < 8) {
    float o = bc[threadIdx.x];
#pragma unroll
    for (int g = 0; g < 4; ++g) o += pool[g]*Wc[threadIdx.x*4 + g];
    out[threadIdx.x] = o;
  }
}

// ---------------------------------------------------------------------------
// Launch. Input order = setup_inputs() dict order with params flattened in
// insertion order of make_params -> 64 arrays.
// ---------------------------------------------------------------------------
extern "C" void kernel_launch(void* const* d_in, const int* in_sizes, int n_in,
                              void* d_out, int out_size, void* d_ws, size_t ws_size,
                              hipStream_t stream) {
  (void)in_sizes; (void)n_in; (void)out_size; (void)ws_size;
  const float* x     = (const float*)d_in[0];
  const float* embed = (const float*)d_in[1];
  auto F = [&](int i) { return (const float*)d_in[i]; };

  const int N = NPTS;
  float* w = (float*)d_ws;
  float* c0_0    = w;            // N      (L0 cat0)
  float* c1_0    = w + 1*N;      // 3N     (L0 cat1)
  float* cat0_1  = w + 4*N;      // 8N
  float* cat1_1  = w + 12*N;     // 36N
  float* cat0_2  = w + 48*N;     // 8N
  float* cat1_2  = w + 56*N;     // 36N
  float* feat0_1 = w + 92*N;     // 4N
  float* feat1_1 = w + 96*N;     // 12N
  float* feat0_2 = w + 108*N;    // 4N
  float* feat1_2 = w + 112*N;    // 12N
  float* feat0_3 = w + 124*N;    // 4N
  float* feat1_3 = w + 128*N;    // 12N   (total 140N floats)

  // zero all atomic accumulation buffers (first 92N floats)
  hipMemsetAsync(w, 0, (size_t)92 * N * sizeof(float), stream);

  // ---- layer 0 ----
  layer0_msg<<<dim3(6, 12), 256, 0, stream>>>(x, embed,
      F(2), F(3), F(4), F(5),      // R00: W1,b1,W2,b2
      F(6), F(7), F(8), F(9),      // R01
      c0_0, c1_0);
  self_interact<<<6, 256, 0, stream>>>(c0_0, c1_0, F(10), F(11), F(12), F(13),
      feat0_1, feat1_1, 1, 1);

  // ---- layer 1 ----
  Radial5 P1;
  for (int p = 0; p < 5; ++p) {
    P1.W1[p] = F(14 + 4*p); P1.b1[p] = F(15 + 4*p);
    P1.W2[p] = F(16 + 4*p); P1.b2[p] = F(17 + 4*p);
  }
  layer_msg_wmma<<<dim3(N, 6), 32, 0, stream>>>(x, feat0_1, feat1_1, P1, cat0_1, cat1_1);
  self_interact<<<6, 256, 0, stream>>>(cat0_1, cat1_1, F(34), F(35), F(36), F(37),
      feat0_2, feat1_2, 8, 12);

  // ---- layer 2 ----
  Radial5 P2;
  for (int p = 0; p < 5; ++p) {
    P2.W1[p] = F(38 + 4*p); P2.b1[p] = F(39 + 4*p);
    P2.W2[p] = F(40 + 4*p); P2.b2[p] = F(41 + 4*p);
  }
  layer_msg_wmma<<<dim3(N, 6), 32, 0, stream>>>(x, feat0_2, feat1_2, P2, cat0_2, cat1_2);
  self_interact<<<6, 256, 0, stream>>>(cat0_2, cat1_2, F(58), F(59), F(60), F(61),
      feat0_3, feat1_3, 8, 12);

  // ---- head ----
  head_kernel<<<1, 256, 0, stream>>>(feat0_3, F(62), F(63), (float*)d_out);
}